// ImagePatch_23081154249439
// MI455X (gfx1250) — compile-verified
//
#include <hip/hip_runtime.h>
#include <stdint.h>

// Problem constants (match reference)
#define Bsz 32
#define Np  64
#define Cch 3
#define Him 512
#define Wim 512
#define Ppatch 16
#define ROWLEN 768   // P*P*3
#define HW ((long)Him * (long)Wim)

typedef float float4v __attribute__((ext_vector_type(4)));

// ---------------- CDNA5 async LDS <-> global helpers (gfx1250 ISA ch.10 async ops) ----
__device__ __forceinline__ void async_load_b128_nt(uint32_t lds_off, const void* gaddr) {
    // copies 16 bytes from the global address gaddr to the LDS offset lds_off, ASYNCcnt-tracked, non-temporal read
    asm volatile("global_load_async_to_lds_b128 %0, %1, off th:TH_LOAD_NT"
                 :: "v"(lds_off), "v"(gaddr) : "memory");
}
__device__ __forceinline__ void async_store_b128(void* gaddr, uint32_t lds_off) {
    // copies 16 bytes from the LDS offset lds_off to the global address gaddr, ASYNCcnt-tracked, RT (keep in L2)
    asm volatile("global_store_async_from_lds_b128 %0, %1, off"
                 :: "v"(gaddr), "v"(lds_off) : "memory");
}
__device__ __forceinline__ void async_wait0() {
    asm volatile("s_wait_asynccnt 0" ::: "memory");
}

// ---------------- Kernel 1: out = image (100.7 MB streaming copy via async LDS path) --
// Each block moves 16 KB chunks: 4 async b128 loads in flight per lane, one drain,
// 4 async stores, one drain. Loads complete in order; load/store completion is
// unordered, so batched full-drain waits are the correct fencing with one counter.
__global__ void copy_image_async_kernel(const float4v* __restrict__ src,
                                        float4v* __restrict__ dst,
                                        long n4) {
    __shared__ float4v stage[4 * 256];       // 16 KB: 4 slots per thread
    const int tid = threadIdx.x;
    // generic LDS pointer low 32 bits == wave-relative LDS byte offset
    const uint32_t loff0 = (uint32_t)(uintptr_t)&stage[tid];
    const uint32_t loff1 = loff0 + 1 * 256 * 16;
    const uint32_t loff2 = loff0 + 2 * 256 * 16;
    const uint32_t loff3 = loff0 + 3 * 256 * 16;

    const long nchunks = n4 >> 10;           // 1024 float4s per chunk
    for (long cb = blockIdx.x; cb < nchunks; cb += gridDim.x) {
        const float4v* s = src + (cb << 10) + tid;
        float4v*       d = dst + (cb << 10) + tid;
        async_load_b128_nt(loff0, (const void*)(s +   0));
        async_load_b128_nt(loff1, (const void*)(s + 256));
        async_load_b128_nt(loff2, (const void*)(s + 512));
        async_load_b128_nt(loff3, (const void*)(s + 768));
        async_wait0();                        // all 4 LDS slots populated
        async_store_b128((void*)(d +   0), loff0);
        async_store_b128((void*)(d + 256), loff1);
        async_store_b128((void*)(d + 512), loff2);
        async_store_b128((void*)(d + 768), loff3);
        async_wait0();                        // stores consumed LDS before slot reuse
    }

    // Tail (n4 % 1024) — zero for this problem size, kept for safety.
    long t = (nchunks << 10) + (long)blockIdx.x * blockDim.x + tid;
    const long stride = (long)gridDim.x * blockDim.x;
    for (; t < n4; t += stride) dst[t] = src[t];
}

// ---------------- Kernel 2: scatter-add patches (atomic: patches overlap) -------------
__global__ void scatter_patches_kernel(const float* __restrict__ emb,
                                       const int*   __restrict__ labels,
                                       const int*   __restrict__ top_left,
                                       float*       __restrict__ out) {
    const int p = blockIdx.x;                // patch id in [0, B*N)
    const int t = threadIdx.x;               // pixel id in [0, 256): t = i*16 + j
    const int i = t >> 4;
    const int j = t & 15;

    const int label = labels[p];
    const int r0 = top_left[2 * p + 0];
    const int c0 = top_left[2 * p + 1];
    const int b  = p >> 6;                   // p / N

    const long pix      = (long)(r0 + i) * Wim + (long)(c0 + j);
    const long out_base = (long)b * Cch * HW + pix;
    const float* erow   = emb + (long)label * ROWLEN + t;   // coalesced: 256 consecutive

    #pragma unroll
    for (int c = 0; c < Cch; ++c) {
        // vals[b,n,i,j,c] = emb[label][c*256 + i*16 + j]
        const float v = erow[c * (Ppatch * Ppatch)];
        // overlapping patches must sum -> fp32 atomic add, non-returning, agent scope
        __hip_atomic_fetch_add(out + out_base + (long)c * HW, v,
                               __ATOMIC_RELAXED, __HIP_MEMORY_SCOPE_AGENT);
    }
}

// ---------------- Launch --------------------------------------------------------------
extern "C" void kernel_launch(void* const* d_in, const int* in_sizes, int n_in,
                              void* d_out, int out_size, void* d_ws, size_t ws_size,
                              hipStream_t stream) {
    const float* image    = (const float*)d_in[0];   // [B,3,512,512] f32
    const float* emb      = (const float*)d_in[1];   // [128,768]     f32
    const int*   labels   = (const int*)  d_in[2];   // [B,N]         i32
    const int*   top_left = (const int*)  d_in[3];   // [B,N,2]       i32
    float*       out      = (float*)d_out;           // [B,3,512,512] f32

    const long n4 = (long)out_size / 4;              // 6,291,456 float4 elements

    // 6144 chunks of 16 KB; 2048 blocks -> 3 chunks each, ~16K waves of DMA parallelism
    copy_image_async_kernel<<<2048, 256, 0, stream>>>(
        (const float4v*)image, (float4v*)out, n4);

    // Scatter-add: one block per patch (B*N = 2048), ordered after copy on same stream
    scatter_patches_kernel<<<Bsz * Np, 256, 0, stream>>>(emb, labels, top_left, out);
}